// QuantumWarpLayer_18829136625876
// MI455X (gfx1250) — compile-verified
//
#include <hip/hip_runtime.h>
#include <hip/hip_bf16.h>
#include <math.h>

// Problem constants
#define BATCH 8
#define DIN   6
#define NH    4
#define OCH   64
#define HWDIM 256
#define SPX   (HWDIM * HWDIM)        // 65536
#define DNH   (DIN * NH)             // 24

typedef __attribute__((ext_vector_type(2))) float v2f;
typedef __attribute__((ext_vector_type(8))) float v8f;

// Async copy helpers (CDNA5 async-to-LDS path; tracked by ASYNCcnt).
// LDS byte address = low 32 bits of a generic pointer to __shared__ memory.
__device__ __forceinline__ void async_ld_b128(unsigned ldsaddr, const void* gaddr) {
  asm volatile("global_load_async_to_lds_b128 %0, %1, off"
               :: "v"(ldsaddr), "v"(gaddr) : "memory");
}
__device__ __forceinline__ void async_ld_b32(unsigned ldsaddr, const void* gaddr) {
  asm volatile("global_load_async_to_lds_b32 %0, %1, off"
               :: "v"(ldsaddr), "v"(gaddr) : "memory");
}
__device__ __forceinline__ void wait_asynccnt0() {
  asm volatile("s_wait_asynccnt 0x0" ::: "memory");
}

// ---------------------------------------------------------------------------
// Kernel 1: quantum-warp attention, one thread per (b, s) pixel.
// cos/sin(phase) are block-invariant -> computed once into LDS.
// Output stored in [b][nh][s][d] flattened order so the torch-style
// transpose+reshape becomes rows of 24 contiguous floats.
// ---------------------------------------------------------------------------
__global__ __launch_bounds__(256)
void attn_kernel(const float* __restrict__ x, const float* __restrict__ phase,
                 const float* __restrict__ qw, const float* __restrict__ qb,
                 const float* __restrict__ kw, const float* __restrict__ kb,
                 const float* __restrict__ vw, const float* __restrict__ vb,
                 float* __restrict__ attn_flat) {
  __shared__ float s_cp[DNH], s_sp[DNH];
  int tt = threadIdx.x;
  if (tt < DNH) {
    float ph = phase[tt];
    s_cp[tt] = cosf(ph);
    s_sp[tt] = sinf(ph);
  }
  __syncthreads();

  int idx = blockIdx.x * blockDim.x + threadIdx.x;   // grid covers exactly B*SPX
  int b = idx >> 16;          // / SPX
  int s = idx & (SPX - 1);

  float xf[DIN];
#pragma unroll
  for (int c = 0; c < DIN; ++c)
    xf[c] = x[((b * DIN + c) << 16) + s];

  float qc[DNH], qs[DNH], kc[DNH], ks[DNH], vv[DNH];
#pragma unroll
  for (int j = 0; j < DNH; ++j) {
    float q = qb[j], k = kb[j], v = vb[j];
#pragma unroll
    for (int c = 0; c < DIN; ++c) {
      q += xf[c] * qw[j * DIN + c];
      k += xf[c] * kw[j * DIN + c];
      v += xf[c] * vw[j * DIN + c];
    }
    float cp = s_cp[j], sp = s_sp[j];
    qc[j] = q * cp; qs[j] = q * sp;
    kc[j] = k * cp; ks[j] = k * sp;
    vv[j] = v;
  }

  const float scale = 0.4082482904638631f;  // 1/sqrt(6)
  int base = b * (NH * SPX * DIN) + s * DIN;
#pragma unroll
  for (int h = 0; h < NH; ++h) {
    float sc[NH];
    float mx = -3.4e38f;
#pragma unroll
    for (int t = 0; t < NH; ++t) {
      float a = 0.f;
#pragma unroll
      for (int d = 0; d < DIN; ++d)
        a += qc[h * DIN + d] * kc[t * DIN + d] + qs[h * DIN + d] * ks[t * DIN + d];
      a *= scale;
      sc[t] = a;
      mx = fmaxf(mx, a);
    }
    float sum = 0.f;
#pragma unroll
    for (int t = 0; t < NH; ++t) { sc[t] = expf(sc[t] - mx); sum += sc[t]; }
    float inv = 1.0f / sum;
#pragma unroll
    for (int d = 0; d < DIN; ++d) {
      float o = 0.f;
#pragma unroll
      for (int t = 0; t < NH; ++t) o += sc[t] * vv[t * DIN + d];
      attn_flat[base + h * (SPX * DIN) + d] = o * inv;
    }
  }
}

// ---------------------------------------------------------------------------
// Kernel 2: warped = rows24(attn_flat) @ ow^T + ob via V_WMMA_F32_16X16X4_F32.
// The 16x24 A tile (1536 contiguous bytes) is shared by all 4 waves: wave 0
// DMAs it to LDS with global_load_async_to_lds_b128, then every wave feeds
// its WMMAs from LDS. Grid = BATCH * (SPX/16), block = 128 (4 waves).
// ---------------------------------------------------------------------------
__global__ __launch_bounds__(128)
void warped_gemm(const float* __restrict__ attn_flat,
                 const float* __restrict__ ow, const float* __restrict__ ob,
                 float* __restrict__ warped) {
  __shared__ float tileA[16 * DNH];   // 1536 B, rows of 24 floats (96 B, 16B-aligned)

  int blk = blockIdx.x;
  int b = blk >> 12;           // / (SPX/16)
  int tile = blk & 4095;
  int row0 = tile << 4;

  // --- async stage A tile: 96 x 16B chunks, wave 0 only ---
  if (threadIdx.x < 32) {
    const char* gA = (const char*)(attn_flat + b * (NH * SPX * DIN) + row0 * DNH);
    unsigned ldsbase = (unsigned)(uintptr_t)&tileA[0];
#pragma unroll
    for (int it = 0; it < 3; ++it) {
      unsigned off = (threadIdx.x + it * 32) * 16;
      async_ld_b128(ldsbase + off, gA + off);
    }
    wait_asynccnt0();
  }
  __syncthreads();

  int wave = threadIdx.x >> 5;
  int lane = threadIdx.x & 31;
  int m = lane & 15;                 // A-matrix row (pixel)
  int kh = (lane >> 4) << 1;         // K sub-offset: 0 for lanes 0-15, 2 for 16-31
  int n0 = wave << 4;
  int nl = lane & 15;                // B/C column (channel)

  const float* Arow = &tileA[m * DNH];
  const float* Brow = ow + (n0 + nl) * DNH;

  v8f acc = {};
#pragma unroll
  for (int kk = 0; kk < 6; ++kk) {
    v2f a, bm;
    a.x = Arow[kk * 4 + kh];
    a.y = Arow[kk * 4 + kh + 1];
    bm.x = Brow[kk * 4 + kh];        // B[k][n] = ow[n*24 + k]
    bm.y = Brow[kk * 4 + kh + 1];
    acc = __builtin_amdgcn_wmma_f32_16x16x4_f32(false, a, false, bm,
                                                (short)0, acc, false, false);
  }

  int o = n0 + nl;
  float bias = ob[o];
  float* W = warped + b * (OCH * SPX) + o * SPX + row0 + ((lane >> 4) << 3);
#pragma unroll
  for (int r = 0; r < 8; ++r) W[r] = acc[r] + bias;
}

// ---------------------------------------------------------------------------
// Kernel 3: fused conv3x3 + BN + Mish -> conv1x1 (WMMA) + sigmoid gate ->
// resonance + combine with warped -> per-pixel LayerNorm over channels.
// x patch staged into LDS with async-to-LDS (borders zeroed via ds_store to
// disjoint addresses). Block = 256 threads per 16-pixel row tile.
// ---------------------------------------------------------------------------
#define FS 68   // padded LDS channel stride (avoid bank conflicts)

__global__ __launch_bounds__(256)
void fused_tail(const float* __restrict__ x, const float* __restrict__ warped,
                const float* __restrict__ c1w, const float* __restrict__ c1b,
                const float* __restrict__ bn_g, const float* __restrict__ bn_b,
                const float* __restrict__ bn_m, const float* __restrict__ bn_v,
                const float* __restrict__ c2w, const float* __restrict__ c2b,
                const float* __restrict__ primes,
                const float* __restrict__ ln_g, const float* __restrict__ ln_b,
                float* __restrict__ out) {
  int blk = blockIdx.x;
  int wtile = blk & 15;              // W/16 = 16
  int hrow  = (blk >> 4) & 255;      // H = 256
  int b     = blk >> 12;             // batch
  int w0    = wtile << 4;
  int t     = threadIdx.x;

  __shared__ float xpatch[DIN][3][18];
  __shared__ float frac[16 * FS];
  __shared__ float ebuf[16 * FS];
  __shared__ float obuf[16 * FS];
  __shared__ float resv[16];
  __shared__ float muv[16], rsv[16];

  // --- stage x patch (6 ch x 3 rows x 18 cols): async loads for in-bounds,
  //     plain zero stores for the zero-padded halo (disjoint addresses) ---
  for (int i = t; i < DIN * 3 * 18; i += 256) {
    int c  = i / 54;
    int rr = (i % 54) / 18;
    int cc = i % 18;
    int hh = hrow + rr - 1;
    int ww = w0 + cc - 1;
    if (hh >= 0 && hh < HWDIM && ww >= 0 && ww < HWDIM) {
      const float* g = &x[((b * DIN + c) << 16) + (hh << 8) + ww];
      async_ld_b32((unsigned)(uintptr_t)&xpatch[c][rr][cc], g);
    } else {
      xpatch[c][rr][cc] = 0.f;
    }
  }
  wait_asynccnt0();
  __syncthreads();

  // --- conv3x3 + BN + Mish -> frac[pixel][channel]; 4 channels per thread ---
  int p  = t & 15;
  int og = (t >> 4) << 2;
#pragma unroll
  for (int i = 0; i < 4; ++i) {
    int o = og + i;
    float acc = c1b[o];
#pragma unroll
    for (int c = 0; c < DIN; ++c)
#pragma unroll
      for (int r3 = 0; r3 < 3; ++r3)
#pragma unroll
        for (int c3 = 0; c3 < 3; ++c3)
          acc += xpatch[c][r3][p + c3] * c1w[((o * DIN + c) * 3 + r3) * 3 + c3];
    float yv = (acc - bn_m[o]) * rsqrtf(bn_v[o] + 1e-5f) * bn_g[o] + bn_b[o];
    float sp = (yv > 20.f) ? yv : log1pf(expf(yv));
    frac[p * FS + o] = yv * tanhf(sp);
  }

  // --- resonance per pixel from staged center row ---
  if (t < 16) {
    float r = 0.f;
#pragma unroll
    for (int c = 0; c < DIN; ++c)
      r += sinf(6.283185307179586f * xpatch[c][1][t + 1] * primes[c]);
    resv[t] = r;
  }
  __syncthreads();

  // --- conv1x1 (64x64) as 16x64x64 GEMM via fp32 WMMA; waves 0-3 ---
  int wave = t >> 5;
  int lane = t & 31;
  if (wave < 4) {
    int m  = lane & 15;
    int kh = (lane >> 4) << 1;
    int n0 = wave << 4;
    int nl = lane & 15;
    const float* Brow = c2w + (n0 + nl) * OCH;   // B[k][n] = c2w[n*64 + k]
    v8f acc = {};
#pragma unroll
    for (int kk = 0; kk < 16; ++kk) {
      v2f a, bm;
      a.x = frac[m * FS + kk * 4 + kh];
      a.y = frac[m * FS + kk * 4 + kh + 1];
      bm.x = Brow[kk * 4 + kh];
      bm.y = Brow[kk * 4 + kh + 1];
      acc = __builtin_amdgcn_wmma_f32_16x16x4_f32(false, a, false, bm,
                                                  (short)0, acc, false, false);
    }
    int o = n0 + nl;
    float cb = c2b[o];
#pragma unroll
    for (int r = 0; r < 8; ++r) {
      int pix = r + ((lane >> 4) << 3);
      float e = acc[r] + cb;
      ebuf[pix * FS + o] = 1.0f / (1.0f + expf(-e));   // sigmoid energy gate
    }
  }
  __syncthreads();

  // --- combine: warped * energy + fractal * (1 + 0.1*resonance) ---
  int sbase = b * (OCH * SPX) + (hrow << 8) + w0 + p;
  float resp = 1.0f + 0.1f * resv[p];
#pragma unroll
  for (int i = 0; i < 4; ++i) {
    int o = og + i;
    float wv = warped[sbase + o * SPX];
    obuf[p * FS + o] = wv * ebuf[p * FS + o] + frac[p * FS + o] * resp;
  }
  __syncthreads();

  // --- LayerNorm over 64 channels, per pixel ---
  if (t < 16) {
    float sm = 0.f;
    for (int o = 0; o < OCH; ++o) sm += obuf[t * FS + o];
    float mean = sm * (1.0f / OCH);
    float vs = 0.f;
    for (int o = 0; o < OCH; ++o) {
      float d = obuf[t * FS + o] - mean;
      vs += d * d;
    }
    muv[t] = mean;
    rsv[t] = rsqrtf(vs * (1.0f / OCH) + 1e-5f);
  }
  __syncthreads();

#pragma unroll
  for (int i = 0; i < 4; ++i) {
    int o = og + i;
    float val = (obuf[p * FS + o] - muv[p]) * rsv[p] * ln_g[o] + ln_b[o];
    out[sbase + o * SPX] = val;
  }
}

// ---------------------------------------------------------------------------
// Launch
// ---------------------------------------------------------------------------
extern "C" void kernel_launch(void* const* d_in, const int* in_sizes, int n_in,
                              void* d_out, int out_size, void* d_ws, size_t ws_size,
                              hipStream_t stream) {
  const float* x      = (const float*)d_in[0];
  const float* phase  = (const float*)d_in[1];
  const float* qw     = (const float*)d_in[2];
  const float* qb     = (const float*)d_in[3];
  const float* kw     = (const float*)d_in[4];
  const float* kb     = (const float*)d_in[5];
  const float* vw     = (const float*)d_in[6];
  const float* vb     = (const float*)d_in[7];
  const float* ow     = (const float*)d_in[8];
  const float* ob     = (const float*)d_in[9];
  const float* c1w    = (const float*)d_in[10];
  const float* c1b    = (const float*)d_in[11];
  const float* bn_g   = (const float*)d_in[12];
  const float* bn_b   = (const float*)d_in[13];
  const float* bn_m   = (const float*)d_in[14];
  const float* bn_v   = (const float*)d_in[15];
  const float* c2w    = (const float*)d_in[16];
  const float* c2b    = (const float*)d_in[17];
  const float* primes = (const float*)d_in[18];
  const float* ln_g   = (const float*)d_in[19];
  const float* ln_b   = (const float*)d_in[20];
  float* outp = (float*)d_out;

  float* ws = (float*)d_ws;
  float* attn_flat = ws;                                   // 12,582,912 floats
  float* warped    = ws + (size_t)BATCH * NH * SPX * DIN;  // 33,554,432 floats

  attn_kernel<<<(BATCH * SPX) / 256, 256, 0, stream>>>(
      x, phase, qw, qb, kw, kb, vw, vb, attn_flat);

  warped_gemm<<<BATCH * (SPX / 16), 128, 0, stream>>>(
      attn_flat, ow, ob, warped);

  fused_tail<<<BATCH * HWDIM * (HWDIM / 16), 256, 0, stream>>>(
      x, warped, c1w, c1b, bn_g, bn_b, bn_m, bn_v, c2w, c2b,
      primes, ln_g, ln_b, outp);
}